// SpectralAngleLoss_83373905149953
// MI455X (gfx1250) — compile-verified
//
#include <hip/hip_runtime.h>

// SpectralAngleLoss for MI455X (gfx1250, wave32).
// One block per sample. CDNA5 paths used:
//  - global_load_async_to_lds_b128 (ASYNCcnt) to stage the 5 input rows,
//    overlapped with zeroing the LDS histograms
//  - ds_add_f32 LDS atomics for the 2x2000-bin scatter
//  - b128 LDS traffic for zero/reduce, wave32 shuffle reduction
// Workload is HBM-bound (40 MB @ 23.3 TB/s ~ 1.7us) with no matmul
// structure, so WMMA is intentionally not used.

#define NBINS  2000          // 500 float4
#define NBIN4  500
#define NPEAK  256
#define INV_PI 0.31830988618379067f

__global__ __launch_bounds__(256) void sal_per_sample(
    const float* __restrict__ pmz,  const float* __restrict__ pint,
    const float* __restrict__ tmz,  const float* __restrict__ tint,
    const float* __restrict__ tmask, float* __restrict__ out_angle)
{
    __shared__ float4 hp4[NBIN4];        // pred histogram  (8000 B)
    __shared__ float4 ht4[NBIN4];        // target histogram (8000 B)
    __shared__ float4 stage4[5 * NPEAK / 4];  // staged rows: [row][peak], 5120 B
    __shared__ float  red[24];           // cross-wave reduction scratch

    float* hp    = (float*)hp4;
    float* ht    = (float*)ht4;
    float* stage = (float*)stage4;

    const int tid = threadIdx.x;
    const size_t rowbase = (size_t)blockIdx.x * NPEAK;

    // ---- Stage 5 rows (5 KB) as 320 x 16B chunks via async b128 copies.
    // Chunk c: row r = c>>6, 16B offset k = c&63. Thread t takes chunk t,
    // threads 0..63 additionally take chunk 256+t (row 4 = mask).
    {
        const int r = tid >> 6;            // 0..3
        const int k = tid & 63;            // 16B chunk within row
        const float* src = (r < 2) ? (r == 0 ? pmz : pint)
                                   : (r == 2 ? tmz : tint);
        const float* g = src + rowbase + k * 4;
        // AS3->generic cast puts the DS byte offset in the low 32 bits,
        // which is the VDST (LDS address) operand of the async load.
        unsigned l = (unsigned)(uintptr_t)stage + (unsigned)(r * 1024 + k * 16);
        asm volatile("global_load_async_to_lds_b128 %0, %1, off"
                     :: "v"(l), "v"(g) : "memory");
        if (tid < 64) {
            const float* g4 = tmask + rowbase + tid * 4;
            unsigned l4 = (unsigned)(uintptr_t)stage + (unsigned)(4 * 1024 + tid * 16);
            asm volatile("global_load_async_to_lds_b128 %0, %1, off"
                         :: "v"(l4), "v"(g4) : "memory");
        }
    }

    // ---- Zero histograms (b128 stores) while the async copies are in flight.
    {
        const float4 z = make_float4(0.f, 0.f, 0.f, 0.f);
        for (int i = tid; i < NBIN4; i += 256) { hp4[i] = z; ht4[i] = z; }
    }

    // Each wave drains its own async writes, then block-barrier publishes
    // both the staged data and the zeroed histograms.
    asm volatile("s_wait_asynccnt 0" ::: "memory");
    __syncthreads();

    // ---- Histogram scatter via LDS float atomics (ds_add_f32).
    {
        float v_pmz = stage[0 * NPEAK + tid];
        float v_pin = stage[1 * NPEAK + tid];
        float v_tmz = stage[2 * NPEAK + tid];
        float v_tin = stage[3 * NPEAK + tid];
        float v_tmk = stage[4 * NPEAK + tid];
        int pb = (int)(v_pmz * 2000.0f);        // trunc toward zero, as reference
        int tb = (int)(v_tmz * 2000.0f);
        pb = min(max(pb, 0), NBINS - 1);
        tb = min(max(tb, 0), NBINS - 1);
        atomicAdd(&hp[pb], v_pin);
        atomicAdd(&ht[tb], v_tin * v_tmk);
    }
    __syncthreads();

    // ---- Triple dot product (p.p, p.t, t.t) over 500 float4 bins.
    float pp = 0.0f, pt = 0.0f, tt = 0.0f;
    for (int i = tid; i < NBIN4; i += 256) {
        float4 p = hp4[i];
        float4 t = ht4[i];
        pp = fmaf(p.x, p.x, fmaf(p.y, p.y, fmaf(p.z, p.z, fmaf(p.w, p.w, pp))));
        pt = fmaf(p.x, t.x, fmaf(p.y, t.y, fmaf(p.z, t.z, fmaf(p.w, t.w, pt))));
        tt = fmaf(t.x, t.x, fmaf(t.y, t.y, fmaf(t.z, t.z, fmaf(t.w, t.w, tt))));
    }

    // wave32 reduction
    #pragma unroll
    for (int off = 16; off > 0; off >>= 1) {
        pp += __shfl_down(pp, off, 32);
        pt += __shfl_down(pt, off, 32);
        tt += __shfl_down(tt, off, 32);
    }
    const int wave = tid >> 5;
    if ((tid & 31) == 0) { red[wave] = pp; red[8 + wave] = pt; red[16 + wave] = tt; }
    __syncthreads();

    if (tid == 0) {
        float spp = 0.0f, spt = 0.0f, stt = 0.0f;
        #pragma unroll
        for (int w = 0; w < 8; ++w) {
            spp += red[w]; spt += red[8 + w]; stt += red[16 + w];
        }
        float pn = fmaxf(sqrtf(spp), 1e-8f);
        float tn = fmaxf(sqrtf(stt), 1e-8f);
        float c  = spt / (pn * tn);
        c = fminf(1.0f, fmaxf(-1.0f, c));
        out_angle[blockIdx.x] = acosf(c) * INV_PI;
    }
}

// Deterministic mean over the per-sample angles.
__global__ __launch_bounds__(256) void sal_reduce(
    const float* __restrict__ angles, float* __restrict__ out,
    int n, float inv_n)
{
    __shared__ float red[256];
    float s = 0.0f;
    for (int i = threadIdx.x; i < n; i += 256) s += angles[i];
    red[threadIdx.x] = s;
    __syncthreads();
    for (int off = 128; off > 0; off >>= 1) {
        if (threadIdx.x < off) red[threadIdx.x] += red[threadIdx.x + off];
        __syncthreads();
    }
    if (threadIdx.x == 0) out[0] = red[0] * inv_n;
}

extern "C" void kernel_launch(void* const* d_in, const int* in_sizes, int n_in,
                              void* d_out, int out_size, void* d_ws, size_t ws_size,
                              hipStream_t stream) {
    const float* pmz   = (const float*)d_in[0];
    const float* pint  = (const float*)d_in[1];
    const float* tmz   = (const float*)d_in[2];
    const float* tint  = (const float*)d_in[3];
    const float* tmask = (const float*)d_in[4];
    float* ws  = (float*)d_ws;
    float* out = (float*)d_out;

    const int B = in_sizes[0] / NPEAK;   // 8192 for the reference shapes

    sal_per_sample<<<B, 256, 0, stream>>>(pmz, pint, tmz, tint, tmask, ws);
    sal_reduce<<<1, 256, 0, stream>>>(ws, out, B, 1.0f / (float)B);
}